// AddPoissonNoise_8907762172166
// MI455X (gfx1250) — compile-verified
//
#include <hip/hip_runtime.h>
#include <hip/hip_bf16.h>

#define AS1 __attribute__((address_space(1)))
#define AS3 __attribute__((address_space(3)))

#ifndef __has_builtin
#define __has_builtin(x) 0
#endif

typedef int   v4i __attribute__((ext_vector_type(4)));
typedef float v4f __attribute__((ext_vector_type(4)));

// ---------------- CDNA5 async global->LDS copy (B128) ----------------
__device__ __forceinline__ void async_load_b128(const int4* gptr, int4* lptr) {
#if __has_builtin(__builtin_amdgcn_global_load_async_to_lds_b128)
    AS1 v4i* gp = (AS1 v4i*)((void*)gptr);   // generic -> global addrspace, drop const
    AS3 v4i* lp = (AS3 v4i*)((void*)lptr);   // generic -> local addrspace
    __builtin_amdgcn_global_load_async_to_lds_b128(gp, lp, 0, 0);
#else
    unsigned lds_addr = (unsigned)(unsigned long long)(const void*)lptr; // low 32b of generic ptr == LDS byte offset
    unsigned long long ga = (unsigned long long)(const void*)gptr;
    asm volatile("global_load_async_to_lds_b128 %0, %1, off"
                 :: "v"(lds_addr), "v"(ga) : "memory");
#endif
}

__device__ __forceinline__ void wait_async0() {
#if __has_builtin(__builtin_amdgcn_s_wait_asynccnt)
    __builtin_amdgcn_s_wait_asynccnt(0);
#else
    asm volatile("s_wait_asynccnt 0" ::: "memory");
#endif
}

// ---------------- stateless counter RNG ----------------
__device__ __forceinline__ unsigned hash32(unsigned x) {
    x ^= x >> 17; x *= 0xed5ad4bbu;
    x ^= x >> 11; x *= 0xac4c1b51u;
    x ^= x >> 15; x *= 0x31848babu;
    x ^= x >> 14;
    return x;
}

struct Rng {
    unsigned s;
    __device__ __forceinline__ unsigned next() {
        s = s * 747796405u + 2891336453u;
        unsigned w = ((s >> ((s >> 28) + 4u)) ^ s) * 277803737u;
        return (w >> 22) ^ w;
    }
    // uniform in (0,1], never 0 -> safe for log
    __device__ __forceinline__ float uf() {
        return (float)(next() >> 8) * 0x1p-24f + 0x1p-25f;
    }
};

// ---------------- Poisson sampler: Knuth for small lambda, normal approx otherwise ----------------
__device__ __forceinline__ float poisson_clip255(int v, unsigned idx) {
    if (v <= 0) return 0.0f;
    Rng r;
    r.s = hash32(idx * 0x9E3779B9u + 0x6C8E9CF5u);
    float lam = (float)v;
    int k;
    if (lam < 12.0f) {
        // Knuth: product of uniforms vs exp(-lambda); expected lam+1 iterations
        float L = __expf(-lam);
        float p = 1.0f;
        k = 0;
        #pragma unroll 1
        do { p *= r.uf(); ++k; } while (p > L && k < 64);
        k -= 1;
    } else {
        // Normal approximation via Box-Muller (valid for lambda >= ~12)
        float u1 = r.uf();
        float u2 = r.uf();
        float mag = __fsqrt_rn(-2.0f * __logf(u1));
        float z   = mag * __cosf(6.28318530718f * u2);
        float nf  = fmaf(__fsqrt_rn(lam), z, lam + 0.5f);
        k = (int)nf;
    }
    k = k < 0 ? 0 : (k > 255 ? 255 : k);
    return (float)k;
}

// ---------------- kernel ----------------
// 256 threads = 8 wave32; each wave stages DEPTH batches of 32 x int4 (16B/lane)
// into wave-private LDS slots via async loads, waits on ASYNCcnt, then computes.
#define DEPTH 4

__global__ void __launch_bounds__(256)
AddPoissonNoise_kernel(const int* __restrict__ in, float* __restrict__ out, int n) {
    __shared__ int4 smem[8 * DEPTH * 32];   // 16 KB / block (of 320 KB per WGP)

    const int tid  = threadIdx.x;
    const int wave = tid >> 5;
    const int lane = tid & 31;
    const int4* in4  = (const int4*)in;
    v4f*        out4 = (v4f*)out;

    const int nvec   = n >> 2;                 // int4 elements
    const int stages = nvec / 256;             // 256 int4 per stage per block
    const int nSuper = stages / DEPTH;         // DEPTH stages per superstage

    for (int q = blockIdx.x; q < nSuper; q += gridDim.x) {
        const int baseVec = q * (DEPTH * 256);

        // Issue DEPTH async B128 loads (wave-private LDS slots) -> deep MLP, no VGPR cost
        #pragma unroll
        for (int d = 0; d < DEPTH; ++d) {
            const int v = baseVec + d * 256 + tid;
            async_load_b128(in4 + v, &smem[wave * (DEPTH * 32) + d * 32 + lane]);
        }
        wait_async0();

        // Consume from LDS (ds_load_b128), compute 4 Poisson draws, vector NT store
        #pragma unroll
        for (int d = 0; d < DEPTH; ++d) {
            const int v = baseVec + d * 256 + tid;
            const int4 x = smem[wave * (DEPTH * 32) + d * 32 + lane];
            const unsigned e = (unsigned)v * 4u;
            v4f rres;
            rres.x = poisson_clip255(x.x, e + 0u);
            rres.y = poisson_clip255(x.y, e + 1u);
            rres.z = poisson_clip255(x.z, e + 2u);
            rres.w = poisson_clip255(x.w, e + 3u);
            // Output stream has zero reuse and exceeds 192MB L2 -> non-temporal store
            __builtin_nontemporal_store(rres, &out4[v]);
        }
        // No block barrier needed: LDS slots are wave-private, and the next batch
        // of async writes is issued only after this wave consumed its reads.
    }

    // Scalar tail (empty for 16x2048x2048, kept for generality)
    const int tailStart = nSuper * DEPTH * 256 * 4;
    for (int i = tailStart + blockIdx.x * blockDim.x + threadIdx.x; i < n;
         i += gridDim.x * blockDim.x) {
        const int v = __builtin_nontemporal_load(&in[i]);
        const float rv = poisson_clip255(v, (unsigned)i);
        __builtin_nontemporal_store(rv, &out[i]);
    }
}

extern "C" void kernel_launch(void* const* d_in, const int* in_sizes, int n_in,
                              void* d_out, int out_size, void* d_ws, size_t ws_size,
                              hipStream_t stream) {
    const int*  img = (const int*)d_in[0];
    float*      out = (float*)d_out;
    const int   n   = in_sizes[0];

    dim3 grid(2048), block(256);
    AddPoissonNoise_kernel<<<grid, block, 0, stream>>>(img, out, n);
}